// MultiTaskReranker_48885317763309
// MI455X (gfx1250) — compile-verified
//
#include <hip/hip_runtime.h>
#include <hip/hip_bf16.h>

typedef __attribute__((ext_vector_type(2))) float v2f;
typedef __attribute__((ext_vector_type(8))) float v8f;

#define NN      100000
#define EE      1600000
#define DIN     256
#define HH      128
#define CC      10

// ---------------------------------------------------------------------------
// Kernel 1: zero nbr_sum + deg in workspace
// ---------------------------------------------------------------------------
__global__ void MultiTaskReranker_zero_kernel(float* __restrict__ p, size_t n) {
    size_t i = (size_t)blockIdx.x * blockDim.x + threadIdx.x;
    const size_t stride = (size_t)gridDim.x * blockDim.x;
    for (; i < n; i += stride) p[i] = 0.0f;
}

// ---------------------------------------------------------------------------
// Kernel 2: edge scatter.  One wave32 per edge: gather x[src] row (256 f32),
// atomically add into nbr_sum[dst].  nbr_sum is 102 MB -> resident in the
// 192 MB L2, so these atomics never leave the chip.
// ---------------------------------------------------------------------------
__global__ __launch_bounds__(256) void MultiTaskReranker_scatter_kernel(
    const long long* __restrict__ src_idx, const long long* __restrict__ dst_idx,
    const float* __restrict__ x, float* __restrict__ nbr_sum,
    float* __restrict__ deg, int num_edges)
{
    const int wave = blockIdx.x * (blockDim.x >> 5) + (threadIdx.x >> 5);
    const int lane = threadIdx.x & 31;
    if (wave >= num_edges) return;
    const long long s = src_idx[wave];
    const long long d = dst_idx[wave];
    if (lane == 0) atomicAdd(&deg[d], 1.0f);
    const float* __restrict__ xs = x + (size_t)s * DIN;
    float* __restrict__ nd = nbr_sum + (size_t)d * DIN;
#pragma unroll
    for (int i = 0; i < 8; ++i) {
        // lanes access consecutive addresses each iteration -> coalesced
        atomicAdd(&nd[lane + 32 * i], xs[lane + 32 * i]);
    }
}

// ---------------------------------------------------------------------------
// Kernel 3: fused SAGEConv + residual + heads for a 16-node tile.
// 256 threads = 8 waves; wave w owns output-column tile [16w, 16w+16).
// f32 WMMA 16x16x4, K accumulated over D_IN=256 (64 steps).
//
// VGPR layouts (CDNA5 ISA 7.12.2):
//   A (16x4 f32):  lane = M + 16*kh, reg0 = K=2*kh, reg1 = K=2*kh+1
//   B (4x16 f32):  lane = N + 16*kh, reg0 = K=2*kh, reg1 = K=2*kh+1
//   C/D (16x16):   reg v, lanes 0-15 -> M=v, lanes 16-31 -> M=v+8; N = lane&15
// ---------------------------------------------------------------------------
__global__ __launch_bounds__(256) void MultiTaskReranker_fused_kernel(
    const float* __restrict__ x, const float* __restrict__ rr,
    const float* __restrict__ W_l, const float* __restrict__ b_l,
    const float* __restrict__ W_r, const float* __restrict__ W_res,
    const float* __restrict__ b_res, const float* __restrict__ W_score,
    const float* __restrict__ b_score, const float* __restrict__ W_crit,
    const float* __restrict__ b_crit, const float* __restrict__ alpha_p,
    const float* __restrict__ nbr_sum, const float* __restrict__ deg,
    float* __restrict__ out_scores, float* __restrict__ out_crit)
{
    __shared__ float s_x[16 * DIN];     // 16 KB
    __shared__ float s_agg[16 * DIN];   // 16 KB
    __shared__ float s_h[16 * HH];      // 8 KB
    __shared__ float s_inv[16];

    const int tid  = threadIdx.x;
    const int base = blockIdx.x * 16;   // first node of this tile (N % 16 == 0)

    if (tid < 16) {
        const float d = deg[base + tid];
        s_inv[tid] = 1.0f / fmaxf(d, 1.0f);
    }
    __syncthreads();

    // Stage x tile and mean-normalized agg tile into LDS (float4 traffic).
    for (int i = tid; i < 16 * DIN / 4; i += 256) {
        const int row = i / (DIN / 4);
        const int c4  = i - row * (DIN / 4);
        const size_t g = (size_t)(base + row) * (DIN / 4) + c4;
        ((float4*)s_x)[i] = ((const float4*)x)[g];
        float4 nv = ((const float4*)nbr_sum)[g];
        const float inv = s_inv[row];
        nv.x *= inv; nv.y *= inv; nv.z *= inv; nv.w *= inv;
        ((float4*)s_agg)[i] = nv;
    }
    __syncthreads();

    const int wave    = tid >> 5;       // 0..7 -> H column tile
    const int lane    = tid & 31;
    const int mn      = lane & 15;      // M index for A, N index for B
    const int kh      = lane >> 4;      // K-half selector
    const int colBase = wave * 16;

    const float* __restrict__ wl_row   = W_l   + (size_t)(colBase + mn) * DIN;
    const float* __restrict__ wr_row   = W_r   + (size_t)(colBase + mn) * DIN;
    const float* __restrict__ wres_row = W_res + (size_t)(colBase + mn) * DIN;

    v8f acc1 = {};   // agg@W_l.T + x@W_r.T  (pre-ReLU)
    v8f acc2 = {};   // x@W_res.T            (residual)

    for (int k0 = 0; k0 < DIN; k0 += 4) {
        const int kk = k0 + 2 * kh;                      // even -> 8B aligned
        const v2f a_agg = *(const v2f*)&s_agg[mn * DIN + kk];
        const v2f a_x   = *(const v2f*)&s_x[mn * DIN + kk];
        const v2f b_wl  = *(const v2f*)&wl_row[kk];
        const v2f b_wr  = *(const v2f*)&wr_row[kk];
        const v2f b_wrs = *(const v2f*)&wres_row[kk];
        acc1 = __builtin_amdgcn_wmma_f32_16x16x4_f32(
            false, a_agg, false, b_wl, (short)0, acc1, false, false);
        acc1 = __builtin_amdgcn_wmma_f32_16x16x4_f32(
            false, a_x, false, b_wr, (short)0, acc1, false, false);
        acc2 = __builtin_amdgcn_wmma_f32_16x16x4_f32(
            false, a_x, false, b_wrs, (short)0, acc2, false, false);
    }

    // Epilogue: h = relu(acc1 + b_l) + acc2 + b_res  ->  LDS
    {
        const float blv   = b_l[colBase + mn];
        const float bresv = b_res[colBase + mn];
#pragma unroll
        for (int v = 0; v < 8; ++v) {
            const int row = v + 8 * kh;                  // node within tile
            const float hv = fmaxf(acc1[v] + blv, 0.0f) + acc2[v] + bresv;
            s_h[row * HH + colBase + mn] = hv;
        }
    }
    __syncthreads();

    // Heads: 16 nodes x (10 criterion cols + 1 score) = 176 dots of length 128
    if (tid < 16 * 11) {
        const int node_m = tid / 11;
        const int o      = tid - node_m * 11;
        const float* __restrict__ hrow = &s_h[node_m * HH];
        float acc = 0.0f;
        if (o < CC) {
            const float* __restrict__ wrow = W_crit + (size_t)o * HH;
#pragma unroll 4
            for (int j = 0; j < HH; ++j) acc += hrow[j] * wrow[j];
            out_crit[(size_t)(base + node_m) * CC + o] = acc + b_crit[o];
        } else {
#pragma unroll 4
            for (int j = 0; j < HH; ++j) acc += hrow[j] * W_score[j];
            const float g  = acc + b_score[0];
            const float al = alpha_p[0];
            const float a  = 1.0f / (1.0f + __expf(-al));   // sigmoid(alpha)
            out_scores[base + node_m] = a * rr[base + node_m] + (1.0f - a) * g;
        }
    }
}

// ---------------------------------------------------------------------------
extern "C" void kernel_launch(void* const* d_in, const int* in_sizes, int n_in,
                              void* d_out, int out_size, void* d_ws, size_t ws_size,
                              hipStream_t stream) {
    const float*     x       = (const float*)d_in[0];
    const long long* eidx    = (const long long*)d_in[1];   // int64 [2, E]
    const float*     rr      = (const float*)d_in[2];
    const float*     W_l     = (const float*)d_in[3];
    const float*     b_l     = (const float*)d_in[4];
    const float*     W_r     = (const float*)d_in[5];
    const float*     W_res   = (const float*)d_in[6];
    const float*     b_res   = (const float*)d_in[7];
    const float*     W_score = (const float*)d_in[8];
    const float*     b_score = (const float*)d_in[9];
    const float*     W_crit  = (const float*)d_in[10];
    const float*     b_crit  = (const float*)d_in[11];
    const float*     alpha   = (const float*)d_in[12];

    // Workspace: nbr_sum [N,256] + deg [N]  (~103 MB)
    float* nbr_sum = (float*)d_ws;
    float* deg     = nbr_sum + (size_t)NN * DIN;

    float* out_scores = (float*)d_out;          // [N]
    float* out_crit   = out_scores + NN;        // [N, 10]

    const size_t zcount = (size_t)NN * DIN + NN;
    MultiTaskReranker_zero_kernel<<<2048, 256, 0, stream>>>(nbr_sum, zcount);

    MultiTaskReranker_scatter_kernel<<<(EE + 7) / 8, 256, 0, stream>>>(
        eidx, eidx + EE, x, nbr_sum, deg, EE);

    MultiTaskReranker_fused_kernel<<<NN / 16, 256, 0, stream>>>(
        x, rr, W_l, b_l, W_r, W_res, b_res, W_score, b_score,
        W_crit, b_crit, alpha, nbr_sum, deg, out_scores, out_crit);
}